// memoryClusterer_66640712564885
// MI455X (gfx1250) — compile-verified
//
#include <hip/hip_runtime.h>
#include <math.h>

// Problem constants from the reference: H=1080, W=1920, S=2000, C=8
constexpr int H  = 1080;
constexpr int W  = 1920;
constexpr int HW = H * W;          // 2,073,600 (divisible by 4)
constexpr int NP = 29;             // 1 preseg plane + 7*4 attmap planes

typedef int   v4i __attribute__((ext_vector_type(4)));
typedef float v4f __attribute__((ext_vector_type(4)));

__device__ __forceinline__ v4f ld_v4f_off(const float* __restrict__ base, unsigned byteOff) {
    return *reinterpret_cast<const v4f*>(reinterpret_cast<const char*>(base) + byteOff);
}

__global__ __launch_bounds__(256, 1)
void memoryClusterer_attmaps_kernel(const int*   __restrict__ spx,
                                    const int*   __restrict__ knn_pred,
                                    const float* __restrict__ knn_dist,
                                    const float* __restrict__ cent_dist,
                                    float*       __restrict__ out)
{
    const int t    = blockIdx.x * blockDim.x + threadIdx.x;  // one float4 of pixels
    const int nvec = HW / 4;
    if (t >= nvec) return;

    // Streamed label read: non-temporal (read once, never again)
    v4i s4 = __builtin_nontemporal_load(reinterpret_cast<const v4i*>(spx) + t);
    // 0-based superpixel index per pixel; labels are 1..S so this is in [0, S)
    unsigned idx[4] = { (unsigned)(s4.x - 1), (unsigned)(s4.y - 1),
                        (unsigned)(s4.z - 1), (unsigned)(s4.w - 1) };

    v4f res[NP];   // 29 output planes, one float per pixel-slot

#pragma unroll
    for (int i = 0; i < 4; ++i) {
        const unsigned id    = idx[i];
        const unsigned rowB  = id * 32u;   // byte offset of 8-float table row
        const unsigned predB = id * 4u;    // byte offset into knn_pred

        // Table gathers: regular temporal hint -> stay hot in WGP$/L2 (136 KB total).
        // 32-bit byte offsets so the backend can use SGPR64-base + VGPR32-offset
        // addressing (GVS) with immediate offsets, instead of per-lane 64-bit adds.
        const float pred = (float)(*reinterpret_cast<const int*>(
            reinterpret_cast<const char*>(knn_pred) + predB));
        const v4f k0 = ld_v4f_off(knn_dist,  rowB);
        const v4f k1 = ld_v4f_off(knn_dist,  rowB + 16u);
        const v4f c0 = ld_v4f_off(cent_dist, rowB);
        const v4f c1 = ld_v4f_off(cent_dist, rowB + 16u);

        const float ka[8] = { k0.x, k0.y, k0.z, k0.w, k1.x, k1.y, k1.z, k1.w };
        const float ca[8] = { c0.x, c0.y, c0.z, c0.w, c1.x, c1.y, c1.z, c1.w };

        // prefix maxes: kp[j] = max(ka[0..j-1])  (j = 1..7)
        float kp[8], cp[8];
        kp[1] = ka[0];  cp[1] = ca[0];
#pragma unroll
        for (int j = 2; j < 8; ++j) {
            kp[j] = fmaxf(kp[j - 1], ka[j - 1]);
            cp[j] = fmaxf(cp[j - 1], ca[j - 1]);
        }
        // suffix maxes: ks[j] = max(ka[j+1..7])  (j = 1..6), ks[7] = -inf
        float ks[8], cs[8];
        ks[7] = -INFINITY;  cs[7] = -INFINITY;
#pragma unroll
        for (int j = 6; j >= 1; --j) {
            ks[j] = fmaxf(ka[j + 1], ks[j + 1]);
            cs[j] = fmaxf(ca[j + 1], cs[j + 1]);
        }

        // Slot results into component i of each plane's float4
        reinterpret_cast<float*>(&res[0])[i] = pred;
#pragma unroll
        for (int j = 1; j < 8; ++j) {
            const int b = 1 + (j - 1) * 4;
            reinterpret_cast<float*>(&res[b + 0])[i] = ka[j];
            reinterpret_cast<float*>(&res[b + 1])[i] = fmaxf(kp[j], ks[j]);
            reinterpret_cast<float*>(&res[b + 2])[i] = ca[j];
            reinterpret_cast<float*>(&res[b + 3])[i] = fmaxf(cp[j], cs[j]);
        }
    }

    // 240 MB output stream: non-temporal b128 stores so we don't evict the
    // gather tables from L2 (TH=NT on global_store_b128)
#pragma unroll
    for (int p = 0; p < NP; ++p) {
        __builtin_nontemporal_store(res[p],
            reinterpret_cast<v4f*>(out + (size_t)p * HW) + t);
    }
}

extern "C" void kernel_launch(void* const* d_in, const int* in_sizes, int n_in,
                              void* d_out, int out_size, void* d_ws, size_t ws_size,
                              hipStream_t stream)
{
    const int*   spx       = (const int*)  d_in[0];  // [1,1,H,W] int32, labels 1..S
    const int*   knn_pred  = (const int*)  d_in[1];  // [S] int32
    const float* knn_dist  = (const float*)d_in[2];  // [S,C] float32
    const float* cent_dist = (const float*)d_in[3];  // [S,C] float32
    float*       out       = (float*)d_out;          // [HW] preseg ++ [7*4*HW] attmaps

    const int nvec    = HW / 4;                      // 518,400 float4 pixel groups
    const int threads = 256;
    const int blocks  = (nvec + threads - 1) / threads;

    hipLaunchKernelGGL(memoryClusterer_attmaps_kernel,
                       dim3(blocks), dim3(threads), 0, stream,
                       spx, knn_pred, knn_dist, cent_dist, out);
}